// chebLSTMCell_72619307041315
// MI455X (gfx1250) — compile-verified
//
#include <hip/hip_runtime.h>

// Problem constants (match reference)
#define N_DIM 16384
#define B_DIM 2
#define H_DIM 16
#define C_DIM 64    // packed RHS width: [x(b0)|x(b1)|h(b0)|h(b1)], 4 groups of 16

// TDM chunking for the L stream
#define KC    128   // K elements per TDM chunk (per wave)
#define KCP   132   // LDS row stride in floats (TDM pads 4 DW per 128 DW row)

typedef __attribute__((ext_vector_type(2))) float v2f;
typedef __attribute__((ext_vector_type(8))) float v8f;

#if defined(__HIP_DEVICE_COMPILE__) && __has_builtin(__builtin_amdgcn_tensor_load_to_lds)
#define USE_TDM 1
#else
#define USE_TDM 0
#endif

// D = A(16x4, fp32) * B(4x16, fp32) + C(16x16, fp32), wave32 WMMA.
__device__ __forceinline__ v8f wmma4(v2f a, v2f b, v8f c) {
    return __builtin_amdgcn_wmma_f32_16x16x4_f32(false, a, false, b, (short)0, c,
                                                 false, false);
}

__device__ __forceinline__ float sigmoidf(float x) {
    return 1.0f / (1.0f + __expf(-x));
}

#if USE_TDM
typedef __attribute__((ext_vector_type(4))) unsigned int u32x4;
typedef __attribute__((ext_vector_type(4))) int i32x4;
typedef __attribute__((ext_vector_type(8))) int i32x8;

// Issue a TDM 2D tile load: 16 rows x KC floats of L (row stride N_DIM floats)
// into LDS at byte offset lds_addr, padding +4 DW per 128-DW row (stride KCP).
__device__ __forceinline__ void tdm_issue(unsigned int lds_addr, unsigned long long gaddr) {
    u32x4 g0;
    g0.x = 1u;                                     // count=1, user descriptor, no gather
    g0.y = lds_addr;                               // LDS byte address
    g0.z = (unsigned int)gaddr;                    // global_addr[31:0]
    g0.w = (unsigned int)((gaddr >> 32) & 0x01FFFFFFu) | (2u << 30); // addr[56:32] | type=2
    i32x8 g1;
    g1[0] = (int)((2u << 16)      // data_size = 4 bytes
                | (1u << 20)      // pad_enable
                | (6u << 22)      // pad_interval: every 128 DWORDs
                | (3u << 25));    // pad_amount: 4 DWORDs  -> LDS row stride 132 floats
    g1[1] = (int)(((unsigned)N_DIM & 0xFFFFu) << 16);                 // tensor_dim0[15:0]
    g1[2] = (int)((((unsigned)N_DIM >> 16) & 0xFFFFu)                 // tensor_dim0[31:16]
                | (((unsigned)N_DIM & 0xFFFFu) << 16));               // tensor_dim1[15:0]
    g1[3] = (int)((((unsigned)N_DIM >> 16) & 0xFFFFu)                 // tensor_dim1[31:16]
                | ((unsigned)KC << 16));                              // tile_dim0 = KC
    g1[4] = 16;                                                       // tile_dim1=16, tile_dim2=0
    g1[5] = N_DIM;                                                    // tensor_dim0_stride lo32
    g1[6] = 0;                                                        // stride hi / dim1_stride lo
    g1[7] = 0;
    i32x4 z4 = {0, 0, 0, 0};
#if defined(__clang_major__) && __clang_major__ >= 23
    i32x8 z8 = {0, 0, 0, 0, 0, 0, 0, 0};
    __builtin_amdgcn_tensor_load_to_lds(g0, g1, z4, z4, z8, 0);
#else
    __builtin_amdgcn_tensor_load_to_lds(g0, g1, z4, z4, 0);
#endif
}
#endif  // USE_TDM

// ---------------------------------------------------------------------------
// Pack T0 = [x(b0)|x(b1)|h(b0)|h(b1)]  ->  [N, 64]
// ---------------------------------------------------------------------------
__global__ __launch_bounds__(256) void pack_t0_kernel(
    const float* __restrict__ x,   // [B, N, 16]
    const float* __restrict__ h,   // [B, N, 16]
    float* __restrict__ T0)        // [N, 64]
{
    int idx = blockIdx.x * blockDim.x + threadIdx.x;   // over N*64
    if (idx >= N_DIM * C_DIM) return;
    int n = idx >> 6;
    int c = idx & 63;
    int grp = c >> 4;        // 0: x b0, 1: x b1, 2: h b0, 3: h b1
    int f = c & 15;
    float v;
    if (grp < 2) v = x[((size_t)grp * N_DIM + n) * 16 + f];
    else         v = h[((size_t)(grp - 2) * N_DIM + n) * 16 + f];
    T0[idx] = v;
}

// ---------------------------------------------------------------------------
// Tout = alpha * (L @ Tin) + beta * Tprev          (L: [N,N], Tin/out: [N,64])
// One block = one 16-row output strip. 4 waves split K.
// TDM double-buffers 16xKC L tiles into LDS per wave; LDS reduction at end.
// ---------------------------------------------------------------------------
__global__ __launch_bounds__(128) void cheb_spmm_kernel(
    const float* __restrict__ L,
    const float* __restrict__ Tin,
    const float* __restrict__ Tprev,
    float* __restrict__ Tout,
    float alpha, float beta)
{
    __shared__ float red[4 * 1024];              // 4 waves x (16x64 tile) = 16 KB
#if USE_TDM
    __shared__ float sA[4 * 2 * 16 * KCP];       // 4 waves x 2 buffers x padded tile
#endif

    const int lane = threadIdx.x & 31;
    const int wave = threadIdx.x >> 5;
    const int row0 = blockIdx.x * 16;            // output strip base row
    const int m    = lane & 15;                  // row within strip (A), col-in-tile (B/C)
    const int kh   = lane >> 4;                  // K half-select (0/1)
    const int kk0  = wave * (N_DIM / 4);         // this wave's K range base

    v8f acc[4];
#pragma unroll
    for (int t = 0; t < 4; ++t)
#pragma unroll
        for (int i = 0; i < 8; ++i) acc[t][i] = 0.0f;

#if USE_TDM
    const int NCHUNK = (N_DIM / 4) / KC;         // 32 chunks per wave
    unsigned int ldsbuf[2];
    ldsbuf[0] = (unsigned int)(uintptr_t)&sA[(wave * 2 + 0) * 16 * KCP];
    ldsbuf[1] = (unsigned int)(uintptr_t)&sA[(wave * 2 + 1) * 16 * KCP];
    unsigned long long gbase =
        (unsigned long long)(uintptr_t)(L + (size_t)row0 * N_DIM + kk0);

    tdm_issue(ldsbuf[0], gbase);
    tdm_issue(ldsbuf[1], gbase + (unsigned long long)KC * 4u);

    for (int c = 0; c < NCHUNK; ++c) {
        // oldest outstanding TDM (chunk c) must be complete
        if (c + 1 < NCHUNK) __builtin_amdgcn_s_wait_tensorcnt(1);
        else                __builtin_amdgcn_s_wait_tensorcnt(0);

        const float* Ab = &sA[(wave * 2 + (c & 1)) * 16 * KCP] + m * KCP + 2 * kh;
        const float* Bb = Tin + (size_t)(kk0 + c * KC + 2 * kh) * C_DIM + m;

#pragma unroll 4
        for (int j = 0; j < KC; j += 4) {
            v2f a = *(const v2f*)(Ab + j);                       // ds_load_b64, conflict-free
            const float* br0 = Bb + (size_t)j * C_DIM;
            const float* br1 = br0 + C_DIM;
            v2f b0 = { br0[0],  br1[0]  };
            v2f b1 = { br0[16], br1[16] };
            v2f b2 = { br0[32], br1[32] };
            v2f b3 = { br0[48], br1[48] };
            acc[0] = wmma4(a, b0, acc[0]);
            acc[1] = wmma4(a, b1, acc[1]);
            acc[2] = wmma4(a, b2, acc[2]);
            acc[3] = wmma4(a, b3, acc[3]);
        }

        // all LDS reads of this buffer retired before the DMA overwrites it
        asm volatile("s_wait_dscnt 0x0" ::: "memory");
        if (c + 2 < NCHUNK)
            tdm_issue(ldsbuf[c & 1], gbase + (unsigned long long)(c + 2) * KC * 4u);
    }
#else
    // Fallback: direct per-lane A loads of L (round-1 path)
    const int kk1 = kk0 + N_DIM / 4;
    const float* Lrow = L + (size_t)(row0 + m) * N_DIM + 2 * kh;
#pragma unroll 2
    for (int kk = kk0; kk < kk1; kk += 4) {
        v2f a = *(const v2f*)(Lrow + kk);
        __builtin_prefetch(Lrow + kk + 256, 0, 0);
        const float* br0 = Tin + (size_t)(kk + 2 * kh) * C_DIM + m;
        const float* br1 = br0 + C_DIM;
        v2f b0 = { br0[0],  br1[0]  };
        v2f b1 = { br0[16], br1[16] };
        v2f b2 = { br0[32], br1[32] };
        v2f b3 = { br0[48], br1[48] };
        acc[0] = wmma4(a, b0, acc[0]);
        acc[1] = wmma4(a, b1, acc[1]);
        acc[2] = wmma4(a, b2, acc[2]);
        acc[3] = wmma4(a, b3, acc[3]);
    }
#endif

    // partials -> LDS
#pragma unroll
    for (int t = 0; t < 4; ++t)
#pragma unroll
        for (int i = 0; i < 8; ++i)
            red[wave * 1024 + (t * 8 + i) * 32 + lane] = acc[t][i];
    __syncthreads();

    // reduce 4 waves, apply alpha/beta, scatter to Tout
    for (int idx = threadIdx.x; idx < 1024; idx += 128) {
        float s = red[idx] + red[1024 + idx] + red[2048 + idx] + red[3072 + idx];
        int t = idx >> 8;
        int i = (idx >> 5) & 7;
        int l = idx & 31;
        int row = row0 + i + 8 * (l >> 4);       // C layout: vgpr i -> rows i, i+8
        int col = t * 16 + (l & 15);
        size_t o = (size_t)row * C_DIM + col;
        Tout[o] = alpha * s + beta * Tprev[o];
    }
}

// ---------------------------------------------------------------------------
// Fused projection + LSTM gates.
// One wave = one (batch, 16-row strip). acc tiles 0..3 are gates i,f,o,g.
// ---------------------------------------------------------------------------
__global__ __launch_bounds__(128) void cheb_lstm_kernel(
    const float* __restrict__ T0, const float* __restrict__ T1,
    const float* __restrict__ T2,
    const float* __restrict__ W1, const float* __restrict__ b1,
    const float* __restrict__ W2, const float* __restrict__ b2,
    const float* __restrict__ c_cur,   // [B, N, 16]
    float* __restrict__ out)           // h_next [B,N,16] ++ c_next [B,N,16]
{
    const int lane  = threadIdx.x & 31;
    const int wave  = threadIdx.x >> 5;
    const int strip = blockIdx.x * 4 + wave;      // 0 .. 2*N/16 - 1
    const int b     = strip >> 10;                // batch
    const int srow  = (strip & 1023) * 16;        // row base
    const int m     = lane & 15;
    const int kh    = lane >> 4;

    v8f acc[4];
#pragma unroll
    for (int t = 0; t < 4; ++t) {
        float bb = b1[t * 16 + m] + b2[t * 16 + m];
#pragma unroll
        for (int i = 0; i < 8; ++i) acc[t][i] = bb;
    }

    const float* Tk[3] = { T0, T1, T2 };
#pragma unroll
    for (int k = 0; k < 3; ++k) {
#pragma unroll
        for (int src = 0; src < 2; ++src) {
            // A: 16x16 slice of Tk at columns (src*32 + b*16)
            const float* A = Tk[k] + (size_t)(srow + m) * C_DIM + (src * 32 + b * 16);
            const float* W = (src == 0 ? W1 : W2) + k * 16 * 64;   // [16, 64]
#pragma unroll
            for (int s = 0; s < 4; ++s) {                          // K steps of 4
                v2f a = *(const v2f*)(A + 4 * s + 2 * kh);
                const float* wr0 = W + (size_t)(4 * s + 2 * kh) * 64 + m;
                const float* wr1 = wr0 + 64;
#pragma unroll
                for (int t = 0; t < 4; ++t) {
                    v2f bb = { wr0[t * 16], wr1[t * 16] };
                    acc[t] = wmma4(a, bb, acc[t]);
                }
            }
        }
    }

    // Gate epilogue: tile0=i, tile1=f, tile2=o, tile3=g
    const size_t HALF = (size_t)B_DIM * N_DIM * H_DIM;
#pragma unroll
    for (int i = 0; i < 8; ++i) {
        int row = srow + i + 8 * kh;
        size_t idx = ((size_t)b * N_DIM + row) * H_DIM + m;
        float gi = sigmoidf(acc[0][i]);
        float gf = sigmoidf(acc[1][i]);
        float go = sigmoidf(acc[2][i]);
        float gg = tanhf(acc[3][i]);
        float cn = gf * c_cur[idx] + gi * gg;
        float hn = go * tanhf(cn);
        out[idx]        = hn;   // h_next
        out[HALF + idx] = cn;   // c_next
    }
}

// ---------------------------------------------------------------------------
extern "C" void kernel_launch(void* const* d_in, const int* in_sizes, int n_in,
                              void* d_out, int out_size, void* d_ws, size_t ws_size,
                              hipStream_t stream) {
    const float* x    = (const float*)d_in[0];   // [B, N, 16]
    const float* Lg   = (const float*)d_in[1];   // [N, N]
    const float* h    = (const float*)d_in[2];   // [B, N, 16]
    const float* c    = (const float*)d_in[3];   // [B, N, 16]
    const float* W1   = (const float*)d_in[4];   // [3, 16, 64]
    const float* b1   = (const float*)d_in[5];   // [64]
    const float* W2   = (const float*)d_in[6];   // [3, 16, 64]
    const float* b2   = (const float*)d_in[7];   // [64]
    float* out = (float*)d_out;

    // Workspace: T0, T1, T2 each [N, 64] fp32 (4 MB each, 12 MB total)
    float* T0 = (float*)d_ws;
    float* T1 = T0 + (size_t)N_DIM * C_DIM;
    float* T2 = T1 + (size_t)N_DIM * C_DIM;

    // 1) pack RHS
    pack_t0_kernel<<<(N_DIM * C_DIM + 255) / 256, 256, 0, stream>>>(x, h, T0);

    // 2) T1 = L @ T0
    cheb_spmm_kernel<<<N_DIM / 16, 128, 0, stream>>>(Lg, T0, T0, T1, 1.0f, 0.0f);

    // 3) T2 = 2 * (L @ T1) - T0
    cheb_spmm_kernel<<<N_DIM / 16, 128, 0, stream>>>(Lg, T1, T0, T2, 2.0f, -1.0f);

    // 4) fused projection + LSTM gates
    cheb_lstm_kernel<<<(B_DIM * N_DIM / 16) / 4, 128, 0, stream>>>(
        T0, T1, T2, W1, b1, W2, b2, c, out);
}